// IoU_81106162418346
// MI455X (gfx1250) — compile-verified
//
#include <hip/hip_runtime.h>
#include <cstdint>

#define N_DET  4096
#define N_HB   8        // 2 heads x 4 images, hb = h*4+b (h=0 clean, h=1 patch)
#define NC     80
#define STRIDE 85

typedef float v2f __attribute__((ext_vector_type(2)));
typedef float v8f __attribute__((ext_vector_type(8)));

// ---------------------------------------------------------------- decode ----
// One wave32 per detection: lanes cooperatively reduce argmax over 80 classes.
__global__ void decode_kernel(const float* __restrict__ clean,
                              const float* __restrict__ patch,
                              float4* __restrict__ boxOff,   // +cls*4096 (NMS)
                              float4* __restrict__ boxN,     // /640 (cross IoU)
                              float* __restrict__ conf,
                              int* __restrict__ clsI,
                              int* __restrict__ validI)
{
    int waveFlat = blockIdx.x * 8 + (threadIdx.x >> 5);   // 0..32767
    int lane = threadIdx.x & 31;
    int hb = waveFlat >> 12;
    int i  = waveFlat & (N_DET - 1);
    int h = hb >> 2, b = hb & 3;
    const float* src = (h ? patch : clean) + (size_t)(b * N_DET + i) * STRIDE;

    float bestS = -__builtin_huge_valf();
    int bestC = 0;
#pragma unroll
    for (int k = 0; k < 3; ++k) {
        int c = lane + 32 * k;
        if (c < NC) {
            float s = src[5 + c];
            if (s > bestS || (s == bestS && c < bestC)) { bestS = s; bestC = c; }
        }
    }
#pragma unroll
    for (int d = 16; d >= 1; d >>= 1) {
        float oS = __shfl_xor(bestS, d);
        int   oC = __shfl_xor(bestC, d);
        if (oS > bestS || (oS == bestS && oC < bestC)) { bestS = oS; bestC = oC; }
    }
    if (lane == 0) {
        float cx = src[0], cy = src[1], w = src[2], hh = src[3], obj = src[4];
        float x1 = cx - 0.5f * w, y1 = cy - 0.5f * hh;
        float x2 = cx + 0.5f * w, y2 = cy + 0.5f * hh;
        float cf = obj * bestS;                    // obj >= 0 -> max commutes
        float th = h ? 0.001f : 0.25f;
        int v = (obj > th) && (cf > th);
        float off = (float)bestC * 4096.0f;
        size_t g = (size_t)hb * N_DET + i;
        boxOff[g] = make_float4(x1 + off, y1 + off, x2 + off, y2 + off);
        const float inv = 1.0f / 640.0f;
        boxN[g] = make_float4(x1 * inv, y1 * inv, x2 * inv, y2 * inv);
        conf[g] = cf; clsI[g] = bestC; validI[g] = v;
    }
}

// ------------------------------------------------------------------ rank ----
// Stable descending rank (matches jnp.argsort(-conf)); also zero-init keep.
__global__ void rank_kernel(const float* __restrict__ conf,
                            int* __restrict__ order,
                            int* __restrict__ keepOrig)
{
    __shared__ float tile[256];
    int hb = blockIdx.y;
    int i = blockIdx.x * 256 + threadIdx.x;
    const float* c = conf + (size_t)hb * N_DET;
    float my = c[i];
    int rank = 0;
    for (int t = 0; t < N_DET / 256; ++t) {
        tile[threadIdx.x] = c[t * 256 + threadIdx.x];
        __syncthreads();
#pragma unroll 8
        for (int k = 0; k < 256; ++k) {
            int j = t * 256 + k;
            float cj = tile[k];
            rank += ((cj > my) | ((cj == my) & (j < i))) ? 1 : 0;
        }
        __syncthreads();
    }
    order[(size_t)hb * N_DET + rank] = i;
    keepOrig[(size_t)hb * N_DET + i] = 0;
}

// ---------------------------------------------------------------- gather ----
__global__ void gather_kernel(const int* __restrict__ order,
                              const float4* __restrict__ boxOff,
                              const int* __restrict__ validI,
                              float4* __restrict__ sBox,
                              int* __restrict__ sValid)
{
    size_t g = (size_t)blockIdx.x * 256 + threadIdx.x;  // 0..32767
    size_t base = g & ~(size_t)(N_DET - 1);             // hb*N
    int s = order[g];
    sBox[g] = boxOff[base + s];
    sValid[g] = validI[base + s];
}

// --------------------------------------------------------------- bitmask ----
// One block (128 thr) per sorted row i: division-free IoU>th, ballot -> bits.
__global__ void mask_kernel(const float4* __restrict__ sBox,
                            unsigned* __restrict__ masks)
{
    int i = blockIdx.x, hb = blockIdx.y;
    int tid = threadIdx.x, lane = tid & 31, wave = tid >> 5;
    const float4* sb = sBox + (size_t)hb * N_DET;
    float4 bi = sb[i];
    float areaA = (bi.z - bi.x) * (bi.w - bi.y);
    unsigned* out = masks + ((size_t)hb * N_DET + i) * 128;
    for (int chunk = 0; chunk < N_DET / 128; ++chunk) {
        int j = chunk * 128 + tid;
        unsigned m = 0;
        if (chunk * 128 + 127 > i) {   // block-uniform: chunk holds some j>i
            float4 bj = sb[j];
            float ix = fminf(bi.z, bj.z) - fmaxf(bi.x, bj.x);
            float iy = fminf(bi.w, bj.w) - fmaxf(bi.y, bj.y);
            float inter = fmaxf(ix, 0.f) * fmaxf(iy, 0.f);
            float areaB = (bj.z - bj.x) * (bj.w - bj.y);
            bool pred = (j > i) & (inter > 0.45f * (areaA + areaB - inter));
            m = __builtin_amdgcn_ballot_w32(pred);
        }
        if (lane == 0) out[chunk * 4 + wave] = m;
    }
}

// -------------------------------------------------------- serial greedy -----
// One wave per (head,image). Suppression state: 4 words/lane in registers
// (128 words = 4096 bits). Next mask row prefetched while current is used.
__global__ void nms_kernel(const unsigned* __restrict__ masks,
                           const int* __restrict__ sValid,
                           const int* __restrict__ order,
                           int* __restrict__ keepOrig)
{
    int hb = blockIdx.x;
    int lane = threadIdx.x;
    const unsigned* mb = masks + (size_t)hb * N_DET * 128;
    const int* vs = sValid + (size_t)hb * N_DET;
    const int* od = order + (size_t)hb * N_DET;
    int* kp = keepOrig + (size_t)hb * N_DET;
    int maxdet = (hb < 4) ? 1000 : 30000;

    unsigned supp[4] = {0u, 0u, 0u, 0u};   // words lane, 32+lane, 64+lane, 96+lane
    int kept = 0;
    unsigned n0 = mb[lane], n1 = mb[32 + lane], n2 = mb[64 + lane], n3 = mb[96 + lane];
#pragma unroll
    for (int q = 0; q < 4; ++q) {          // i in [1024q,1024q+1024): bit lives in supp[q]
        for (int t = 0; t < 1024; ++t) {
            int i = q * 1024 + t;
            unsigned c0 = n0, c1 = n1, c2 = n2, c3 = n3;
            int inext = (i + 1 < N_DET) ? i + 1 : i;
            const unsigned* rw = mb + (size_t)inext * 128;   // prefetch next row
            n0 = rw[lane]; n1 = rw[32 + lane]; n2 = rw[64 + lane]; n3 = rw[96 + lane];
            unsigned sw = __shfl(supp[q], (i >> 5) & 31);
            bool active = vs[i] && !((sw >> (i & 31)) & 1u); // wave-uniform
            if (active) {
                ++kept;
                if (lane == 0 && kept <= maxdet) kp[od[i]] = 1;
                supp[0] |= c0; supp[1] |= c1; supp[2] |= c2; supp[3] |= c3;
            }
        }
    }
}

// ------------------------------------------------------ cross IoU + WMMA ----
// Wave handles a 16(clean) x 16(patch) tile per step. Area-sum term of the
// IoU union is a rank-2 outer sum -> V_WMMA_F32_16X16X4_F32:
//   A[m][:] = [area_c[m], 1, 0, 0],  B[:][n] = [1, area_p[n], 0, 0]^T
// D layout: lane = n + 16*(m>=8), VGPR r = m%8  -> per-lane column, 8 rows.
// Inner loop is branchless: v_rcp_f32 instead of IEEE div, cndmask masking.
__global__ void cross_kernel(const float4* __restrict__ boxN,
                             const int* __restrict__ clsI,
                             const int* __restrict__ keepOrig,
                             float* __restrict__ partials)
{
    int tid = threadIdx.x, lane = tid & 31, wave = tid >> 5;
    int b = blockIdx.y;
    int grp = blockIdx.x * 8 + wave;     // 0..255 -> 16 clean rows each
    int m0 = grp * 16;
    int half = lane >> 4, n = lane & 15;

    const float4* cb = boxN + (size_t)b * N_DET;
    const int* ccl = clsI + (size_t)b * N_DET;
    const int* ckp = keepOrig + (size_t)b * N_DET;
    const float4* pb = boxN + (size_t)(4 + b) * N_DET;
    const int* pcl = clsI + (size_t)(4 + b) * N_DET;
    const int* pkp = keepOrig + (size_t)(4 + b) * N_DET;

    float4 cbox = cb[m0 + n];
    int ccls = ccl[m0 + n];
    int ckeep = ckp[m0 + n];
    float areaC = (cbox.z - cbox.x) * (cbox.w - cbox.y);

    // A matrix (16x4 f32): lanes 0-15 -> K0,K1 ; lanes 16-31 -> K2,K3 = 0
    v2f A; A.x = (lane < 16) ? areaC : 0.f; A.y = (lane < 16) ? 1.f : 0.f;

    // Hoist this lane's 8 clean rows (m = r + 8*half) out of the j loop.
    float rx1[8], ry1[8], rx2[8], ry2[8]; int rcl[8], rkp[8];
#pragma unroll
    for (int r = 0; r < 8; ++r) {
        int src = r + half * 8;
        rx1[r] = __shfl(cbox.x, src);  ry1[r] = __shfl(cbox.y, src);
        rx2[r] = __shfl(cbox.z, src);  ry2[r] = __shfl(cbox.w, src);
        rcl[r] = __shfl(ccls, src);    rkp[r] = __shfl(ckeep, src);
    }

    float tm[8];
#pragma unroll
    for (int r = 0; r < 8; ++r) tm[r] = 0.f;

    for (int j0 = 0; j0 < N_DET; j0 += 16) {
        float4 pbox = pb[j0 + n];
        int pc = pcl[j0 + n];
        int pk = pkp[j0 + n];
        float areaP = (pbox.z - pbox.x) * (pbox.w - pbox.y);
        v2f B; B.x = (lane < 16) ? 1.f : 0.f; B.y = (lane < 16) ? areaP : 0.f;
        v8f C = {};
        v8f S = __builtin_amdgcn_wmma_f32_16x16x4_f32(
            false, A, false, B, (short)0, C, false, false);
#pragma unroll
        for (int r = 0; r < 8; ++r) {
            float ix = fminf(rx2[r], pbox.z) - fmaxf(rx1[r], pbox.x);
            float iy = fminf(ry2[r], pbox.w) - fmaxf(ry1[r], pbox.y);
            float inter = fmaxf(ix, 0.f) * fmaxf(iy, 0.f);
            float uni = S[r] - inter;              // area_c[m]+area_p[n] via WMMA
            float iou = inter * __builtin_amdgcn_rcpf(uni);  // v_rcp_f32, no div expand
            bool ok = (pk != 0) & (pc == rcl[r]);            // branchless mask
            tm[r] = fmaxf(tm[r], ok ? iou : 0.f);
        }
    }
    // max over the 16 patch columns: xor butterfly stays within each half.
#pragma unroll
    for (int r = 0; r < 8; ++r) {
#pragma unroll
        for (int d = 1; d < 16; d <<= 1)
            tm[r] = fmaxf(tm[r], __shfl_xor(tm[r], d));
    }
    float num = 0.f;
#pragma unroll
    for (int r = 0; r < 8; ++r) num += (rkp[r] != 0) ? tm[r] : 0.f;
    if ((lane & 15) == 0)
        partials[((size_t)b * 256 + grp) * 2 + half] = num;  // fixed slot
}

// -------------------------------------------------------------- finalize ----
__global__ void finalize_kernel(const float* __restrict__ partials,
                                const int* __restrict__ keepOrig,
                                float* __restrict__ out)
{
    __shared__ float sn[256], sd[256];
    int tid = threadIdx.x;
    float n = 0.f, d = 0.f;
    for (int k = tid; k < 2048; k += 256) n += partials[k];
    for (int k = tid; k < 4 * N_DET; k += 256) d += (float)keepOrig[k]; // clean heads
    sn[tid] = n; sd[tid] = d;
    __syncthreads();
    for (int s = 128; s > 0; s >>= 1) {
        if (tid < s) { sn[tid] += sn[tid + s]; sd[tid] += sd[tid + s]; }
        __syncthreads();
    }
    if (tid == 0) {
        float den = sd[0], num = sn[0];
        out[0] = (den > 0.f) ? (1.f - num / fmaxf(den, 1.f)) : 1.f;
    }
}

// ---------------------------------------------------------------- launch ----
extern "C" void kernel_launch(void* const* d_in, const int* in_sizes, int n_in,
                              void* d_out, int out_size, void* d_ws, size_t ws_size,
                              hipStream_t stream)
{
    (void)in_sizes; (void)n_in; (void)out_size; (void)ws_size;
    const float* clean = (const float*)d_in[0];
    const float* patch = (const float*)d_in[1];

    char* p = (char*)d_ws;
    auto alloc = [&](size_t bytes) -> void* {
        void* r = (void*)p;
        p += (bytes + 255) & ~(size_t)255;
        return r;
    };
    float4*   boxOff   = (float4*)  alloc(sizeof(float4)  * N_HB * N_DET);
    float4*   boxN     = (float4*)  alloc(sizeof(float4)  * N_HB * N_DET);
    float*    conf     = (float*)   alloc(sizeof(float)   * N_HB * N_DET);
    int*      clsI     = (int*)     alloc(sizeof(int)     * N_HB * N_DET);
    int*      validI   = (int*)     alloc(sizeof(int)     * N_HB * N_DET);
    int*      order    = (int*)     alloc(sizeof(int)     * N_HB * N_DET);
    int*      keep     = (int*)     alloc(sizeof(int)     * N_HB * N_DET);
    float4*   sBox     = (float4*)  alloc(sizeof(float4)  * N_HB * N_DET);
    int*      sValid   = (int*)     alloc(sizeof(int)     * N_HB * N_DET);
    unsigned* masks    = (unsigned*)alloc(sizeof(unsigned) * (size_t)N_HB * N_DET * 128);
    float*    partials = (float*)   alloc(sizeof(float) * 2048);

    decode_kernel  <<<dim3(4096),    256, 0, stream>>>(clean, patch, boxOff, boxN,
                                                       conf, clsI, validI);
    rank_kernel    <<<dim3(16, 8),   256, 0, stream>>>(conf, order, keep);
    gather_kernel  <<<dim3(128),     256, 0, stream>>>(order, boxOff, validI, sBox, sValid);
    mask_kernel    <<<dim3(4096, 8), 128, 0, stream>>>(sBox, masks);
    nms_kernel     <<<dim3(8),        32, 0, stream>>>(masks, sValid, order, keep);
    cross_kernel   <<<dim3(32, 4),   256, 0, stream>>>(boxN, clsI, keep, partials);
    finalize_kernel<<<dim3(1),       256, 0, stream>>>(partials, keep, (float*)d_out);
}